// GraphFlow_78374563217497
// MI455X (gfx1250) — compile-verified
//
#include <hip/hip_runtime.h>
#include <cstdint>
#include <cstddef>

#define WAVE 32
#define WPB  2               // waves per block
#define BLK  (WAVE * WPB)    // 64 threads
#define FPF  8               // flows
#define MAT  9               // 3x3 floats
#define LPT  (FPF * MAT)     // 72 floats per thread per array (r or v)
#define FPAD 76              // padded floats per lane region (bank stride 12 -> <=2-way)
#define ZSTG (WAVE * 3)      // 96 floats of z-output staging per wave

// per-wave LDS float offsets
#define OFFV    (WAVE * FPAD)        // 2432: V region
#define OFFZ    (2 * WAVE * FPAD)    // 4864: z-output staging (96)
#define OFFQ    (OFFZ + ZSTG)        // 4960: Q staging, 32*9 = 288 floats (stride 9 -> conflict-free)
#define OFFD    (OFFQ + WAVE * MAT)  // 5248: D staging, 7 WMMAs * 256 floats = 1792
#define OFFZERO (OFFD + 7 * 256)     // 7040: zero word for invalid operand slots
#define WSTRIDE (OFFZERO + 4)        // 7044 floats/wave = 28176 B (16B-aligned)

struct f3 { float x, y, z; };

typedef float v2f __attribute__((ext_vector_type(2)));
typedef float v8f __attribute__((ext_vector_type(8)));

#if defined(__gfx1250__) && __has_builtin(__builtin_amdgcn_wmma_f32_16x16x4_f32)
#define HAVE_WMMA4 1
#else
#define HAVE_WMMA4 0
#endif

// ---- CDNA5 async global<->LDS (ASYNCcnt path) -----------------------------
__device__ __forceinline__ void async_load_b128(unsigned lds_byte_addr, const void* gaddr) {
  asm volatile("global_load_async_to_lds_b128 %0, %1, off"
               :: "v"(lds_byte_addr), "v"(gaddr) : "memory");
}
__device__ __forceinline__ void async_store_b128(void* gaddr, unsigned lds_byte_addr) {
  asm volatile("global_store_async_from_lds_b128 %0, %1, off"
               :: "v"(gaddr), "v"(lds_byte_addr) : "memory");
}
__device__ __forceinline__ void wait_async0() {
#if __has_builtin(__builtin_amdgcn_s_wait_asynccnt)
  __builtin_amdgcn_s_wait_asynccnt(0);
#else
  asm volatile("s_wait_asynccnt 0" ::: "memory");
#endif
}
__device__ __forceinline__ void wait_ds0() {
#if __has_builtin(__builtin_amdgcn_s_wait_dscnt)
  __builtin_amdgcn_s_wait_dscnt(0);
#else
  asm volatile("s_wait_dscnt 0" ::: "memory");
#endif
}
__device__ __forceinline__ void wavebar() {
#if __has_builtin(__builtin_amdgcn_wave_barrier)
  __builtin_amdgcn_wave_barrier();
#else
  asm volatile("" ::: "memory");
#endif
}

__global__ __launch_bounds__(BLK) void graphflow_kernel(
    const float* __restrict__ z_i,   // [B,3]
    const float* __restrict__ r,     // [B,F,3,3]
    const float* __restrict__ v,     // [B,F,3,3]
    const float* __restrict__ bvec,  // [B,F,3]
    const float* __restrict__ alpha, // [B,F]
    const float* __restrict__ beta,  // [B,F]
    const float* __restrict__ z0p,   // [B,F,3]
    const float* __restrict__ w_p,   // [B,F,3]
    const float* __restrict__ v_p,   // [B,F,3]
    const float* __restrict__ b_p,   // [B,F]
    float* __restrict__ out,         // [B,3] then [B]
    int B)
{
  __shared__ float smem[WPB * WSTRIDE]; // 56352 bytes

  const int lane = (int)(threadIdx.x & (WAVE - 1));
  const int wv   = (int)(threadIdx.x / WAVE);
  const int bw0  = (int)blockIdx.x * BLK + wv * WAVE;
  const int b    = bw0 + lane;

  // Low 32 bits of a generic pointer to __shared__ == LDS byte offset.
  const unsigned ldsBase = (unsigned)(size_t)(const void*)smem;
  const unsigned regRb = ldsBase + (unsigned)(wv * WSTRIDE) * 4u;
  const unsigned regVb = regRb + (unsigned)OFFV * 4u;
  const unsigned regZb = regRb + (unsigned)OFFZ * 4u;

  const int baseI = wv * WSTRIDE;      // float index of this wave's region
  const int QI    = baseI + OFFQ;
  const int DI    = baseI + OFFD;
  const int zeroI = baseI + OFFZERO;

  // ---- Stage the wave's contiguous r/v blocks into LDS (coalesced b128) ----
  const char* gR = (const char*)(r + (size_t)bw0 * LPT);
  const char* gV = (const char*)(v + (size_t)bw0 * LPT);
#pragma unroll
  for (int t = 0; t < 18; ++t) {
    unsigned c   = (unsigned)(t * WAVE + lane);
    unsigned lo  = c / 18u;
    unsigned rem = c - lo * 18u;
    unsigned off = lo * (FPAD * 4u) + rem * 16u;
    async_load_b128(regRb + off, gR + (size_t)c * 16u);
    async_load_b128(regVb + off, gV + (size_t)c * 16u);
  }

  // ---- Preload per-thread contiguous scalars while async copies fly --------
  const float4 al0 = *(const float4*)(alpha + (size_t)b * 8 + 0);
  const float4 al1 = *(const float4*)(alpha + (size_t)b * 8 + 4);
  const float4 be0 = *(const float4*)(beta  + (size_t)b * 8 + 0);
  const float4 be1 = *(const float4*)(beta  + (size_t)b * 8 + 4);
  const float4 bp0 = *(const float4*)(b_p   + (size_t)b * 8 + 0);
  const float4 bp1 = *(const float4*)(b_p   + (size_t)b * 8 + 4);
  float aAll[8]  = {al0.x, al0.y, al0.z, al0.w, al1.x, al1.y, al1.z, al1.w};
  float btAll[8] = {be0.x, be0.y, be0.z, be0.w, be1.x, be1.y, be1.z, be1.w};
  float bpAll[8] = {bp0.x, bp0.y, bp0.z, bp0.w, bp1.x, bp1.y, bp1.z, bp1.w};

  const f3 zi = *(const f3*)(z_i + (size_t)b * 3);
  float za = zi.x, zb = zi.y, zc = zi.z;
  float dld = 0.0f;

  smem[zeroI] = 0.0f;                 // zero word for invalid WMMA operand slots

  // Per-lane WMMA slot decomposition (static across f):
  // lane = 16*half + rw ; rw = 3*m5 + cs  (cs = row index for A, col index for B)
  const int half = lane >> 4;
  const int rw   = lane & 15;
  const int m5   = rw / 3;            // block slot within a WMMA (0..4; 5 => pad row)
  const int cs   = rw - m5 * 3;
  const int k0   = 2 * half;          // K of operand VGPR0 ; VGPR1 -> K = 1 + 2*half
  const bool s0ok = (rw < 15);                 // K in {0,2} always < 3
  const bool s1ok = (rw < 15) && (half == 0);  // K=1 ok, K=3 must be zero

  wait_async0();

  const int RlI = baseI + lane * FPAD;          // own R matrix (raw, diag fixed in-place)
  const int VlI = baseI + OFFV + lane * FPAD;

#pragma unroll 1
  for (int f = 0; f < FPF; ++f) {
    const size_t bf = (size_t)b * FPF + f;
    const int o = f * MAT;

    const float V00 = smem[VlI+o+0], V01 = smem[VlI+o+1], V02 = smem[VlI+o+2];
    const float V10 = smem[VlI+o+3], V11 = smem[VlI+o+4], V12 = smem[VlI+o+5];
    const float V20 = smem[VlI+o+6], V21 = smem[VlI+o+7], V22 = smem[VlI+o+8];

    // vectorized per-(b,f) 3-vectors
    const f3 bk = *(const f3*)(bvec + bf * 3);
    const f3 wp = *(const f3*)(w_p  + bf * 3);
    const f3 vp = *(const f3*)(v_p  + bf * 3);
    const f3 zk = *(const f3*)(z0p  + bf * 3);

    // diag fix (in LDS, so WMMA B-gathers see fixed values) + base log-det
    const float d0 = fabsf(smem[RlI+o+0]) + 0.1f;
    const float d1 = fabsf(smem[RlI+o+4]) + 0.1f;
    const float d2 = fabsf(smem[RlI+o+8]) + 0.1f;
    smem[RlI+o+0] = d0; smem[RlI+o+4] = d1; smem[RlI+o+8] = d2;
    dld += logf(0.1f + d0) + logf(0.1f + d1) + logf(0.1f + d2);

    // Q = H0 = I - 2 v0 v0^T
    float Q00 = 1.0f - 2.0f*V00*V00, Q01 = -2.0f*V00*V01, Q02 = -2.0f*V00*V02;
    float Q10 = -2.0f*V01*V00, Q11 = 1.0f - 2.0f*V01*V01, Q12 = -2.0f*V01*V02;
    float Q20 = -2.0f*V02*V00, Q21 = -2.0f*V02*V01, Q22 = 1.0f - 2.0f*V02*V02;
    {  // Q = Q * H1
      const float t0 = Q00*V10 + Q01*V11 + Q02*V12;
      const float t1 = Q10*V10 + Q11*V11 + Q12*V12;
      const float t2 = Q20*V10 + Q21*V11 + Q22*V12;
      Q00 -= 2.0f*t0*V10; Q01 -= 2.0f*t0*V11; Q02 -= 2.0f*t0*V12;
      Q10 -= 2.0f*t1*V10; Q11 -= 2.0f*t1*V11; Q12 -= 2.0f*t1*V12;
      Q20 -= 2.0f*t2*V10; Q21 -= 2.0f*t2*V11; Q22 -= 2.0f*t2*V12;
    }
    {  // Q = Q * H2
      const float t0 = Q00*V20 + Q01*V21 + Q02*V22;
      const float t1 = Q10*V20 + Q11*V21 + Q12*V22;
      const float t2 = Q20*V20 + Q21*V21 + Q22*V22;
      Q00 -= 2.0f*t0*V20; Q01 -= 2.0f*t0*V21; Q02 -= 2.0f*t0*V22;
      Q10 -= 2.0f*t1*V20; Q11 -= 2.0f*t1*V21; Q12 -= 2.0f*t1*V22;
      Q20 -= 2.0f*t2*V20; Q21 -= 2.0f*t2*V21; Q22 -= 2.0f*t2*V22;
    }

    float W00, W01, W02, W10, W11, W12, W20, W21, W22;

#if HAVE_WMMA4
    // ---- W = Q @ R via V_WMMA_F32_16X16X4_F32, 5 block-diagonal 3x3s each ---
    {  // publish Q (stride 9 -> bank-conflict-free)
      const int qb = QI + lane * MAT;
      smem[qb+0]=Q00; smem[qb+1]=Q01; smem[qb+2]=Q02;
      smem[qb+3]=Q10; smem[qb+4]=Q11; smem[qb+5]=Q12;
      smem[qb+6]=Q20; smem[qb+7]=Q21; smem[qb+8]=Q22;
    }
    wavebar();  // same-wave DS ops are in-order; barrier stops compiler reordering
#pragma unroll
    for (int g = 0; g < 7; ++g) {
      const bool gok = (g < 6) | (m5 < 2);       // last WMMA holds matrices 30,31 only
      // A[row=rw, K] = Q_{5g+m5}[cs][K]   (K = k0 for VGPR0, 1 for VGPR1/half0)
      const int ia0 = (s0ok && gok) ? (QI + g*45 + m5*MAT + cs*3 + k0) : zeroI;
      const int ia1 = (s1ok && gok) ? (QI + g*45 + m5*MAT + cs*3 + 1)  : zeroI;
      // B[K, col=rw] = R_{5g+m5}[K][cs]  (R row-major at k*3+cs, diag already fixed)
      const int ib0 = (s0ok && gok) ? (baseI + (g*5 + m5)*FPAD + o + k0*3 + cs) : zeroI;
      const int ib1 = (s1ok && gok) ? (baseI + (g*5 + m5)*FPAD + o + 3 + cs)    : zeroI;
      v2f Aop; Aop.x = smem[ia0]; Aop.y = smem[ia1];
      v2f Bop; Bop.x = smem[ib0]; Bop.y = smem[ib1];
      v8f acc = {};
      acc = __builtin_amdgcn_wmma_f32_16x16x4_f32(false, Aop, false, Bop,
                                                  (short)0, acc, false, false);
      const int db = DI + g * 256 + lane;
#pragma unroll
      for (int j = 0; j < 8; ++j) smem[db + j * 32] = acc[j];
    }
    wavebar();
    {  // gather own W block: matrix index == lane, block j5 of WMMA g
      const int gg = lane / 5, j5 = lane - gg * 5;
      const int db = DI + gg * 256 + 3 * j5;       // + column N = 3*j5 + cj
      const int M0 = 3 * j5, M1 = M0 + 1, M2 = M0 + 2;
      const int r0 = (M0 & 7) * 32 + (M0 >> 3) * 16;
      const int r1 = (M1 & 7) * 32 + (M1 >> 3) * 16;
      const int r2 = (M2 & 7) * 32 + (M2 >> 3) * 16;
      W00 = smem[db+r0+0]; W01 = smem[db+r0+1]; W02 = smem[db+r0+2];
      W10 = smem[db+r1+0]; W11 = smem[db+r1+1]; W12 = smem[db+r1+2];
      W20 = smem[db+r2+0]; W21 = smem[db+r2+1]; W22 = smem[db+r2+2];
    }
#else
    // ---- scalar fallback: W = Q @ R --------------------------------------
    {
      const float R01 = smem[RlI+o+1], R02 = smem[RlI+o+2];
      const float R10 = smem[RlI+o+3], R12 = smem[RlI+o+5];
      const float R20 = smem[RlI+o+6], R21 = smem[RlI+o+7];
      W00 = Q00*d0  + Q01*R10 + Q02*R20;
      W01 = Q00*R01 + Q01*d1  + Q02*R21;
      W02 = Q00*R02 + Q01*R12 + Q02*d2;
      W10 = Q10*d0  + Q11*R10 + Q12*R20;
      W11 = Q10*R01 + Q11*d1  + Q12*R21;
      W12 = Q10*R02 + Q11*R12 + Q12*d2;
      W20 = Q20*d0  + Q21*R10 + Q22*R20;
      W21 = Q20*R01 + Q21*d1  + Q22*R21;
      W22 = Q20*R02 + Q21*R12 + Q22*d2;
    }
#endif

    // z = z @ W + b
    float nz0 = za*W00 + zb*W10 + zc*W20 + bk.x;
    float nz1 = za*W01 + zb*W11 + zc*W21 + bk.y;
    float nz2 = za*W02 + zb*W12 + zc*W22 + bk.z;

    // planar flow
    const float bp  = bpAll[f];
    const float inner = wp.x*nz0 + wp.y*nz1 + wp.z*nz2 + bp;
    const float th = tanhf(inner);
    nz0 += vp.x*th; nz1 += vp.y*th; nz2 += vp.z*th;
    const float inner_post = wp.x*nz0 + wp.y*nz1 + wp.z*nz2 + bp;
    const float thp = tanhf(inner_post);
    const float vw  = vp.x*wp.x + vp.y*wp.y + vp.z*wp.z;
    dld += logf(1.0f + (1.0f - thp*thp) * vw);

    // radial flow
    const float a  = fabsf(aAll[f]);
    const float be = fabsf(btAll[f]) - a;
    const float c0 = nz0 - zk.x;
    const float c1 = nz1 - zk.y;
    const float c2 = nz2 - zk.z;
    const float rz  = sqrtf(c0*c0 + c1*c1 + c2*c2);
    const float den = a + rz;
    const float coef = (den == 0.0f) ? 0.0f : (be / den);
    nz0 += coef*c0; nz1 += coef*c1; nz2 += coef*c2;
    const float den2 = den * den;
    const float term = (den2 == 0.0f) ? 0.0f : ((a * be) / den2);
    const float opc  = 1.0f + coef;
    dld += logf((1.0f + term) * opc * opc);

    za = nz0; zb = nz1; zc = nz2;
  }

  // ---- Coalesced z output via async store-from-LDS -------------------------
  float* zstage = &smem[baseI + OFFZ];
  zstage[lane * 3 + 0] = za;
  zstage[lane * 3 + 1] = zb;
  zstage[lane * 3 + 2] = zc;
  wait_ds0();
  if (lane < 24) {                  // 96 floats = 24 x 16B chunks
    char* gZ = (char*)(out + (size_t)bw0 * 3) + (size_t)lane * 16u;
    async_store_b128(gZ, regZb + (unsigned)lane * 16u);
  }
  // S_ENDPGM performs an implicit wait-idle, covering ASYNCcnt completion.

  out[(size_t)B * 3 + b] = dld;
}

extern "C" void kernel_launch(void* const* d_in, const int* in_sizes, int n_in,
                              void* d_out, int out_size, void* d_ws, size_t ws_size,
                              hipStream_t stream) {
  (void)n_in; (void)out_size; (void)d_ws; (void)ws_size;
  const float* z_i   = (const float*)d_in[0];
  const float* r     = (const float*)d_in[1];
  const float* v     = (const float*)d_in[2];
  const float* bvec  = (const float*)d_in[3];
  const float* alpha = (const float*)d_in[4];
  const float* beta  = (const float*)d_in[5];
  const float* z0p   = (const float*)d_in[6];
  const float* w_p   = (const float*)d_in[7];
  const float* v_p   = (const float*)d_in[8];
  const float* b_p   = (const float*)d_in[9];
  float* out = (float*)d_out;

  const int B = in_sizes[0] / 3;            // 524288; divisible by BLK
  dim3 grid((unsigned)((B + BLK - 1) / BLK));
  dim3 block(BLK);
  graphflow_kernel<<<grid, block, 0, stream>>>(z_i, r, v, bvec, alpha, beta,
                                               z0p, w_p, v_p, b_p, out, B);
}